// DCN_layer_80831284511343
// MI455X (gfx1250) — compile-verified
//
#include <hip/hip_runtime.h>
#include <hip/hip_bf16.h>
#include <math.h>

// ---------------------------------------------------------------------------
// DCNv2 (modulated deformable conv) for MI455X / gfx1250, wave32 + f32 WMMA.
//   B=4, C=64, H=W=128, O=64, K=3x3, stride=1, pad=1, dil=1, dg=1
//
// v2: coalesced patch staging, weight B-fragments streamed from L2 (not LDS)
//     -> kernel1 LDS 104 KB (3 WG/WGP), kernel2 LDS 148 KB (2 WG/WGP).
// ---------------------------------------------------------------------------

typedef __attribute__((ext_vector_type(2))) float v2f;
typedef __attribute__((ext_vector_type(4))) float v4f;
typedef __attribute__((ext_vector_type(8))) float v8f;

#define Bn   4
#define Cc   64
#define Hh   128
#define Ww   128
#define Oo   64
#define K2   9

// ----- kernel 1 (offset/mask conv) geometry -----
#define C1     128            // concat channels
#define N1     27             // offset-conv out channels
#define K1     1152           // 9 * 128  (k = tap*128 + c)
#define S1     288            // K1 / 4
#define PXW    66             // patch width  (64 + halo 2)
#define CSTR   132            // padded channel stride (132 % 64 == 4 -> conflict-free b64)

// ----- kernel 2 (main deformable GEMM) geometry -----
#define Km     576            // 9 * 64   (k = tap*64 + c)
#define Sm     144            // Km / 4
#define KPAD   580            // padded K stride (580 % 64 == 4)

// ---------------------------------------------------------------------------
// Prep: pre-swizzle both weight tensors into per-lane WMMA B-fragment order.
//   frag[t][s][lane][0..1]:
//     lane <16 : { Bmat[4s+0][16t+lane],     Bmat[4s+1][16t+lane]     }
//     lane>=16 : { Bmat[4s+2][16t+lane-16],  Bmat[4s+3][16t+lane-16]  }
// wf1: Bmat[k][n] = w_om[n][c][tap], k = tap*128 + c   (N padded 27->32)
// wf2: Bmat[k][n] = weight[n][c][tap], k = tap*64 + c
// ---------------------------------------------------------------------------
__global__ void k_prep_wfrag(const float* __restrict__ w_om,
                             const float* __restrict__ weight,
                             float* __restrict__ wf1,
                             float* __restrict__ wf2) {
    int i = blockIdx.x * blockDim.x + threadIdx.x;
    if (i < 2 * S1 * 64) {                 // 36864 elements
        int t    = i / (S1 * 64);
        int r    = i - t * (S1 * 64);
        int s    = r >> 6;
        int j    = r & 63;
        int lane = j >> 1;
        int a2   = j & 1;
        int l16  = lane & 15;
        int lh   = lane >> 4;
        int k    = 4 * s + 2 * lh + a2;    // 0..1151
        int n    = 16 * t + l16;           // 0..31
        float v = 0.0f;
        if (n < N1) {
            int tap = k >> 7;              // k / 128
            int c   = k & 127;
            v = w_om[(n * C1 + c) * K2 + tap];
        }
        wf1[i] = v;
    }
    if (i < 4 * Sm * 64) {                 // 36864 elements
        int t    = i / (Sm * 64);
        int r    = i - t * (Sm * 64);
        int s    = r >> 6;
        int j    = r & 63;
        int lane = j >> 1;
        int a2   = j & 1;
        int l16  = lane & 15;
        int lh   = lane >> 4;
        int k    = 4 * s + 2 * lh + a2;    // 0..575
        int n    = 16 * t + l16;           // 0..63
        int tap  = k >> 6;                 // k / 64
        int c    = k & 63;
        wf2[i] = weight[(n * Cc + c) * K2 + tap];
    }
}

// ---------------------------------------------------------------------------
// Kernel 1: om = conv3x3(concat(x, inter), w_om) + b_om ; sigmoid on mask chans.
// Grid: 1024 blocks (b * 128 rows * 2 x-halves), 128 threads (4 waves).
// Each wave: 16 pixels x 32 channels (2 WMMA N-tiles), K = 1152 in steps of 4.
// LDS: patch only (104544 B) -> up to 3 workgroups per WGP.
// ---------------------------------------------------------------------------
__global__ void k_offset_conv(const float* __restrict__ xin,
                              const float* __restrict__ inter,
                              const float* __restrict__ wf1,
                              const float* __restrict__ b_om,
                              float* __restrict__ om) {
    extern __shared__ float smem[];
    float* patch = smem;                         // [3][66][CSTR]  (dy, ix, c)

    const int gid   = blockIdx.x;
    const int b     = gid >> 8;
    const int rem   = gid & 255;
    const int y     = rem >> 1;
    const int xbase = (rem & 1) * 64;
    const int tid   = threadIdx.x;

    // stage input patch, COALESCED: iterate (dy, c, ix) so consecutive threads
    // read consecutive x (contiguous in NCHW); store channel-innermost in LDS.
    for (int i = tid; i < 3 * C1 * PXW; i += 128) {
        int ix = i % PXW;
        int r  = i / PXW;                // dy*128 + c
        int c  = r & 127;
        int dy = r >> 7;
        int gy = y - 1 + dy;
        int gx = xbase - 1 + ix;
        float v = 0.0f;
        if (gy >= 0 && gy < Hh && gx >= 0 && gx < Ww) {
            const float* src = (c < Cc) ? xin : inter;
            int cc = c & 63;
            v = src[((b * Cc + cc) * Hh + gy) * Ww + gx];
        }
        patch[(dy * PXW + ix) * CSTR + c] = v;
    }
    __syncthreads();

    const int wave  = tid >> 5;
    const int lane  = tid & 31;
    const int lhalf = lane >> 4;
    const int l16   = lane & 15;
    const int pxo   = wave * 16;          // wave's pixel offset inside block tile

    // B fragments streamed straight from L2 (identical for every workgroup)
    const v2f* __restrict__ bf0 = (const v2f*)(wf1) + lane;            // [S1][32]
    const v2f* __restrict__ bf1 = (const v2f*)(wf1 + S1 * 64) + lane;

    v8f acc0 = {};
    v8f acc1 = {};
    #pragma unroll 4
    for (int s = 0; s < S1; ++s) {
        int k4  = 4 * s;
        int tap = k4 >> 7;                // uniform: group of 4 never crosses a tap
        int c   = (k4 & 127) + 2 * lhalf;
        int dy  = tap / 3;
        int dx  = tap - 3 * dy;
        // A fragment: 16x4 f32, one ds_load_b64 per lane
        const v2f a  = *(const v2f*)&patch[(dy * PXW + pxo + l16 + dx) * CSTR + c];
        const v2f b0 = bf0[s * 32];
        const v2f b1 = bf1[s * 32];
        acc0 = __builtin_amdgcn_wmma_f32_16x16x4_f32(false, a, false, b0,
                                                     (short)0, acc0, false, false);
        acc1 = __builtin_amdgcn_wmma_f32_16x16x4_f32(false, a, false, b1,
                                                     (short)0, acc1, false, false);
    }

    // epilogue: D vgpr r -> (m = r + 8*lhalf, n = l16 + 16*tile)
    #pragma unroll
    for (int r = 0; r < 8; ++r) {
        int m  = r + 8 * lhalf;
        int gx = xbase + pxo + m;
        {   // tile 0: channels 0..15
            int ch  = l16;
            float v = acc0[r] + b_om[ch];
            om[((b * N1 + ch) * Hh + y) * Ww + gx] = v;
        }
        {   // tile 1: channels 16..31 (valid < 27); 18..26 are mask -> sigmoid
            int ch = 16 + l16;
            if (ch < N1) {
                float v = acc1[r] + b_om[ch];
                if (ch >= 18) v = 1.0f / (1.0f + __expf(-v));
                om[((b * N1 + ch) * Hh + y) * Ww + gx] = v;
            }
        }
    }
}

// ---------------------------------------------------------------------------
// Kernel 2: deformable bilinear sampling into an LDS im2col tile, then
// out[b][o][y][x] = sum_k samp[pix][k] * weight_frag + bias.
// Grid: 1024 blocks, 128 threads (4 waves). Wave: 16 px x 64 out-ch (4 tiles).
// LDS: samp only (148480 B) -> 2 workgroups per WGP (8 waves for gather hiding).
// ---------------------------------------------------------------------------
__global__ void k_dcn_main(const float* __restrict__ xin,
                           const float* __restrict__ om,
                           const float* __restrict__ wf2,
                           const float* __restrict__ bias,
                           float* __restrict__ out) {
    extern __shared__ float smem[];
    float* samp = smem;                    // [64][KPAD]

    const int gid   = blockIdx.x;
    const int b     = gid >> 8;
    const int rem   = gid & 255;
    const int y     = rem >> 1;
    const int xbase = (rem & 1) * 64;
    const int tid   = threadIdx.x;

    // build the deformable-sampled A tile: (pixel, tap) pairs, channel loop inside
    const float* xb = xin + (size_t)b * Cc * Hh * Ww;
    for (int p = tid; p < 64 * K2; p += 128) {
        int pix = p / K2;
        int tap = p - pix * K2;
        int gx  = xbase + pix;
        size_t omb = ((size_t)b * N1 * Hh + (size_t)y) * Ww + gx;
        float oy = om[omb + (size_t)(tap)      * Hh * Ww];
        float ox = om[omb + (size_t)(9  + tap) * Hh * Ww];
        float mk = om[omb + (size_t)(18 + tap) * Hh * Ww];   // sigmoid applied in k1
        int ky = tap / 3;
        int kx = tap - 3 * ky;
        float py  = (float)y  - 1.0f + (float)ky + oy;
        float pxf = (float)gx - 1.0f + (float)kx + ox;
        float y0f = floorf(py), x0f = floorf(pxf);
        float wy = py - y0f, wx = pxf - x0f;
        int y0 = (int)y0f, x0 = (int)x0f;
        int y1 = y0 + 1,   x1 = x0 + 1;
        bool vy0 = (y0 >= 0) & (y0 < Hh);
        bool vy1 = (y1 >= 0) & (y1 < Hh);
        bool vx0 = (x0 >= 0) & (x0 < Ww);
        bool vx1 = (x1 >= 0) & (x1 < Ww);
        float w00 = (1.0f - wy) * (1.0f - wx) * mk; if (!(vy0 && vx0)) w00 = 0.0f;
        float w01 = (1.0f - wy) * wx          * mk; if (!(vy0 && vx1)) w01 = 0.0f;
        float w10 = wy          * (1.0f - wx) * mk; if (!(vy1 && vx0)) w10 = 0.0f;
        float w11 = wy          * wx          * mk; if (!(vy1 && vx1)) w11 = 0.0f;
        int y0c = y0 < 0 ? 0 : (y0 > Hh - 1 ? Hh - 1 : y0);
        int y1c = y1 < 0 ? 0 : (y1 > Hh - 1 ? Hh - 1 : y1);
        int x0c = x0 < 0 ? 0 : (x0 > Ww - 1 ? Ww - 1 : x0);
        int x1c = x1 < 0 ? 0 : (x1 > Ww - 1 ? Ww - 1 : x1);
        int i00 = y0c * Ww + x0c;
        int i01 = y0c * Ww + x1c;
        int i10 = y1c * Ww + x0c;
        int i11 = y1c * Ww + x1c;
        float* dst = &samp[pix * KPAD + tap * Cc];
        // 4 channels at a time: 16 independent gathers in flight, b128 LDS store
        for (int c = 0; c < Cc; c += 4) {
            v4f v;
            #pragma unroll
            for (int u = 0; u < 4; ++u) {
                const float* pch = xb + (size_t)(c + u) * (Hh * Ww);
                v[u] = w00 * pch[i00] + w01 * pch[i01]
                     + w10 * pch[i10] + w11 * pch[i11];
            }
            *(v4f*)&dst[c] = v;
        }
    }
    __syncthreads();

    const int wave  = tid >> 5;
    const int lane  = tid & 31;
    const int lhalf = lane >> 4;
    const int l16   = lane & 15;
    const int pxo   = wave * 16;

    // B fragments streamed straight from L2 (identical for every workgroup)
    const v2f* __restrict__ bfp = (const v2f*)(wf2) + lane;   // [4][Sm][32]

    v8f acc[4] = {{}, {}, {}, {}};
    #pragma unroll 4
    for (int s = 0; s < Sm; ++s) {
        const v2f a = *(const v2f*)&samp[(pxo + l16) * KPAD + 4 * s + 2 * lhalf];
        #pragma unroll
        for (int t = 0; t < 4; ++t) {
            const v2f bf = bfp[(t * Sm + s) * 32];
            acc[t] = __builtin_amdgcn_wmma_f32_16x16x4_f32(false, a, false, bf,
                                                           (short)0, acc[t], false, false);
        }
    }

    #pragma unroll
    for (int t = 0; t < 4; ++t) {
        int ch = t * 16 + l16;
        float bv = bias[ch];
        #pragma unroll
        for (int r = 0; r < 8; ++r) {
            int m  = r + 8 * lhalf;
            int gx = xbase + pxo + m;
            out[((b * Oo + ch) * Hh + y) * Ww + gx] = acc[t][r] + bv;
        }
    }
}

// ---------------------------------------------------------------------------
extern "C" void kernel_launch(void* const* d_in, const int* in_sizes, int n_in,
                              void* d_out, int out_size, void* d_ws, size_t ws_size,
                              hipStream_t stream) {
    const float* xin    = (const float*)d_in[0];   // (4,64,128,128)
    const float* inter  = (const float*)d_in[1];   // (4,64,128,128)
    const float* weight = (const float*)d_in[2];   // (64,64,3,3)
    const float* bias   = (const float*)d_in[3];   // (64,)
    const float* w_om   = (const float*)d_in[4];   // (27,128,3,3)
    const float* b_om   = (const float*)d_in[5];   // (27,)
    float* out = (float*)d_out;                    // (4,64,128,128)

    float* ws  = (float*)d_ws;
    float* om  = ws;                               // 4*27*128*128 = 1769472 floats
    float* wf1 = ws + 1769472;                     // 2*288*64     =   36864 floats
    float* wf2 = wf1 + 36864;                      // 4*144*64     =   36864 floats

    k_prep_wfrag<<<144, 256, 0, stream>>>(w_om, weight, wf1, wf2);

    size_t lds1 = (size_t)(3 * PXW * CSTR) * sizeof(float);        // 104544 B
    k_offset_conv<<<Bn * Hh * 2, 128, lds1, stream>>>(xin, inter, wf1, b_om, om);

    size_t lds2 = (size_t)(64 * KPAD) * sizeof(float);             // 148480 B
    k_dcn_main<<<Bn * Hh * 2, 128, lds2, stream>>>(xin, om, wf2, bias, out);
}